// BA3TGCN2_32684701122592
// MI455X (gfx1250) — compile-verified
//
#include <hip/hip_runtime.h>
#include <math.h>

// ---------------- problem constants (from reference) ----------------
#define BB   16
#define NN   10000
#define CINC 8
#define COUTC 32
#define TP   16        // 2*P time slices
#define EE   160000
#define TRAINP 1.0f    // TRAIN_OR_PREDICT

typedef __attribute__((ext_vector_type(2))) float v2f;
typedef __attribute__((ext_vector_type(8))) float v8f;

// workspace layout (floats):
//   S     [B*N*TP*CIN]  = 20,480,000   aggregated features, layout (b,n,t,c)
//   deg   [N]                          degree -> overwritten with D^-1/2
//   fused [592]                        Mz[256] cz[32] Mh[256] ch[32] ws[16]
#define S_ELEMS   ((size_t)BB * NN * TP * CINC)
#define FZ_M   0
#define FZ_C   256
#define FH_M   288
#define FH_C   544
#define F_WS   576

__device__ __forceinline__ float fast_rcp(float x) {
  return __builtin_amdgcn_rcpf(x);   // v_rcp_f32, ~1 ulp
}

// ---- fold Wc@Wl_top, biases, softmax(attention) --------------------
__global__ void prep_kernel(const float* __restrict__ Wcz, const float* __restrict__ bcz,
                            const float* __restrict__ Wlz, const float* __restrict__ blz,
                            const float* __restrict__ Wch, const float* __restrict__ bch,
                            const float* __restrict__ Wlh, const float* __restrict__ blh,
                            const float* __restrict__ att, float* __restrict__ fused) {
  int tid = threadIdx.x;            // 256 threads
  int c = tid >> 5, o = tid & 31;   // c in [0,8), o in [0,32)
  float sz = 0.f, sh = 0.f;
  for (int k = 0; k < COUTC; ++k) {
    sz += Wcz[c * COUTC + k] * Wlz[k * COUTC + o];   // top half of Wl (k < 32)
    sh += Wch[c * COUTC + k] * Wlh[k * COUTC + o];
  }
  fused[FZ_M + c * COUTC + o] = sz;
  fused[FH_M + c * COUTC + o] = sh;
  if (tid < COUTC) {
    float bz = blz[tid], bh = blh[tid];
    for (int k = 0; k < COUTC; ++k) {
      bz += bcz[k] * Wlz[k * COUTC + tid];
      bh += bch[k] * Wlh[k * COUTC + tid];
    }
    fused[FZ_C + tid] = bz;
    fused[FH_C + tid] = bh;
  }
  if (tid == 0) {
    float m = att[0];
    for (int t = 1; t < TP; ++t) m = fmaxf(m, att[t]);
    float e[TP]; float s = 0.f;
    for (int t = 0; t < TP; ++t) { e[t] = __expf(att[t] - m); s += e[t]; }
    float rs = 1.0f / s;           // host-side-rare path, exactness not critical
    for (int t = 0; t < TP; ++t)
      fused[F_WS + t] = (e[t] * rs) * (t < TP / 2 ? 1.0f : TRAINP);
  }
}

// ---- GCN norm: deg = 1 (self loop) + sum(w over dst); then D^-1/2 ---
__global__ void deg_init_kernel(float* __restrict__ deg) {
  int n = blockIdx.x * 256 + threadIdx.x;
  if (n < NN) deg[n] = 1.0f;
}
__global__ void deg_acc_kernel(const int* __restrict__ eidx, const float* __restrict__ w,
                               float* __restrict__ deg) {
  int e = blockIdx.x * 256 + threadIdx.x;
  if (e < EE) atomicAdd(&deg[eidx[EE + e]], w[e]);
}
__global__ void deg_inv_kernel(float* __restrict__ deg) {
  int n = blockIdx.x * 256 + threadIdx.x;
  if (n < NN) { float d = deg[n]; deg[n] = d > 0.f ? rsqrtf(d) : 0.f; }
}

// ---- S init: self-loop term, transposing (c,t)->(t,c) --------------
__global__ void init_S_kernel(const float* __restrict__ X, const float* __restrict__ dinv,
                              float* __restrict__ S) {
  size_t i = (size_t)blockIdx.x * 256 + threadIdx.x;
  if (i >= S_ELEMS) return;
  int c = (int)(i & 7);
  int t = (int)((i >> 3) & 15);
  size_t bn = i >> 7;                 // b*N + n
  int n = (int)(bn % NN);
  float dv = dinv[n];
  S[i] = dv * dv * X[bn * 128 + (size_t)c * 16 + t];
}

// ---- edge scatter: S[b,dst,t,c] += nu * X[b,src,c,t] ----------------
__global__ void scatter_kernel(const float* __restrict__ X, const int* __restrict__ eidx,
                               const float* __restrict__ w, const float* __restrict__ dinv,
                               float* __restrict__ S) {
  int idx = blockIdx.x * 256 + threadIdx.x;
  if (idx >= EE * BB) return;
  int e = idx % EE;
  int b = idx / EE;
  int s = eidx[e], d = eidx[EE + e];
  float nu = dinv[s] * w[e] * dinv[d];
  size_t bx = ((size_t)b * NN + s) * 128;   // X base (b,src)
  size_t bs = ((size_t)b * NN + d) * 128;   // S base (b,dst)
  for (int c = 0; c < CINC; ++c) {
    const float* xp = X + bx + (size_t)c * 16;
#pragma unroll
    for (int t = 0; t < TP; ++t)
      atomicAdd(&S[bs + (size_t)t * 8 + c], nu * xp[t]);
  }
}

// ---- WMMA gate kernel: one wave per (b,n) tile ----------------------
// Rows of the 16x16 tile are the 16 time slices; K=8 channels split into
// two V_WMMA_F32_16X16X4_F32; N=32 output channels as two 16-col tiles;
// two gates (z,h) -> 8 WMMAs per tile, then exp/rcp-based sigmoid/tanh
// and an attention-weighted reduce over t (8 per half-wave + shfl_xor 16).
__global__ void gate_kernel(const float* __restrict__ S, const float* __restrict__ fused,
                            float* __restrict__ out) {
  const int lane = threadIdx.x & 31;
  const int q = lane & 15;       // row (t) for A / col for B,C,D
  const int h = lane >> 4;       // half-wave selector
  const int wave = blockIdx.x * (blockDim.x >> 5) + (threadIdx.x >> 5);
  const int nwaves = gridDim.x * (blockDim.x >> 5);

  const float* Mz = fused + FZ_M;
  const float* cz = fused + FZ_C;
  const float* Mh = fused + FH_M;
  const float* ch = fused + FH_C;
  const float* wsv = fused + F_WS;

  // hoist B operands (constant across tiles)
  v2f bzlo[2], bzhi[2], bhlo[2], bhhi[2];
  float czv[2], chv[2], wst[8];
#pragma unroll
  for (int j = 0; j < 2; ++j) {
    int col = j * 16 + q;
    v2f t;
    t.x = Mz[(2 * h + 0) * COUTC + col]; t.y = Mz[(2 * h + 1) * COUTC + col]; bzlo[j] = t;
    t.x = Mz[(4 + 2 * h) * COUTC + col]; t.y = Mz[(5 + 2 * h) * COUTC + col]; bzhi[j] = t;
    t.x = Mh[(2 * h + 0) * COUTC + col]; t.y = Mh[(2 * h + 1) * COUTC + col]; bhlo[j] = t;
    t.x = Mh[(4 + 2 * h) * COUTC + col]; t.y = Mh[(5 + 2 * h) * COUTC + col]; bhhi[j] = t;
    czv[j] = cz[col];
    chv[j] = ch[col];
  }
#pragma unroll
  for (int v = 0; v < 8; ++v) wst[v] = wsv[v + 8 * h];

  for (int tile = wave; tile < BB * NN; tile += nwaves) {
    // A operands: row t = q, channels {2h,2h+1} and {4+2h,5+2h}
    size_t base = (size_t)tile * 128 + (size_t)q * 8;
    v2f alo, ahi;
    alo.x = S[base + 2 * h];     alo.y = S[base + 2 * h + 1];
    ahi.x = S[base + 4 + 2 * h]; ahi.y = S[base + 5 + 2 * h];

    v8f az0 = {}, az1 = {}, ah0 = {}, ah1 = {};
    az0 = __builtin_amdgcn_wmma_f32_16x16x4_f32(false, alo, false, bzlo[0], (short)0, az0, false, false);
    az0 = __builtin_amdgcn_wmma_f32_16x16x4_f32(false, ahi, false, bzhi[0], (short)0, az0, false, false);
    az1 = __builtin_amdgcn_wmma_f32_16x16x4_f32(false, alo, false, bzlo[1], (short)0, az1, false, false);
    az1 = __builtin_amdgcn_wmma_f32_16x16x4_f32(false, ahi, false, bzhi[1], (short)0, az1, false, false);
    ah0 = __builtin_amdgcn_wmma_f32_16x16x4_f32(false, alo, false, bhlo[0], (short)0, ah0, false, false);
    ah0 = __builtin_amdgcn_wmma_f32_16x16x4_f32(false, ahi, false, bhhi[0], (short)0, ah0, false, false);
    ah1 = __builtin_amdgcn_wmma_f32_16x16x4_f32(false, alo, false, bhlo[1], (short)0, ah1, false, false);
    ah1 = __builtin_amdgcn_wmma_f32_16x16x4_f32(false, ahi, false, bhhi[1], (short)0, ah1, false, false);

    // epilogue: D[v] holds (t = v + 8h, col = j*16 + q)
    float p0 = 0.f, p1 = 0.f;
#pragma unroll
    for (int v = 0; v < 8; ++v) {
      float yz0 = az0[v] + czv[0];
      float yh0 = ah0[v] + chv[0];
      float yz1 = az1[v] + czv[1];
      float yh1 = ah1[v] + chv[1];
      // (1 - sigmoid(x)) = rcp(1+exp(x));  tanh(x) = 1 - 2*rcp(exp(2x)+1)
      float g0 = fast_rcp(1.f + __expf(yz0));
      float g1 = fast_rcp(1.f + __expf(yz1));
      float t0 = fmaf(-2.f, fast_rcp(__expf(2.f * yh0) + 1.f), 1.f);
      float t1 = fmaf(-2.f, fast_rcp(__expf(2.f * yh1) + 1.f), 1.f);
      p0 = fmaf(wst[v], g0 * t0, p0);
      p1 = fmaf(wst[v], g1 * t1, p1);
    }
    p0 += __shfl_xor(p0, 16, 32);   // combine t<8 and t>=8 halves
    p1 += __shfl_xor(p1, 16, 32);
    if (lane < 16) {
      size_t ob = (size_t)tile * COUTC;
      out[ob + q] = p0;
      out[ob + 16 + q] = p1;
    }
  }
}

extern "C" void kernel_launch(void* const* d_in, const int* in_sizes, int n_in,
                              void* d_out, int out_size, void* d_ws, size_t ws_size,
                              hipStream_t stream) {
  const float* X   = (const float*)d_in[0];
  const int*   ei  = (const int*)d_in[1];
  const float* ew  = (const float*)d_in[2];
  const float* att = (const float*)d_in[3];
  const float* Wcz = (const float*)d_in[4];
  const float* bcz = (const float*)d_in[5];
  const float* Wlz = (const float*)d_in[6];
  const float* blz = (const float*)d_in[7];
  // inputs 8..11 (r gate) are provably dead: H0 == 0 => H0*R == 0
  const float* Wch = (const float*)d_in[12];
  const float* bch = (const float*)d_in[13];
  const float* Wlh = (const float*)d_in[14];
  const float* blh = (const float*)d_in[15];
  float* out = (float*)d_out;

  float* S     = (float*)d_ws;
  float* deg   = S + S_ELEMS;     // N floats, becomes D^-1/2 in place
  float* fused = deg + NN;        // 592 floats

  prep_kernel<<<1, 256, 0, stream>>>(Wcz, bcz, Wlz, blz, Wch, bch, Wlh, blh, att, fused);
  deg_init_kernel<<<(NN + 255) / 256, 256, 0, stream>>>(deg);
  deg_acc_kernel<<<(EE + 255) / 256, 256, 0, stream>>>(ei, ew, deg);
  deg_inv_kernel<<<(NN + 255) / 256, 256, 0, stream>>>(deg);

  const int initBlocks = (int)((S_ELEMS + 255) / 256);
  init_S_kernel<<<initBlocks, 256, 0, stream>>>(X, deg, S);

  scatter_kernel<<<(EE * BB + 255) / 256, 256, 0, stream>>>(X, ei, ew, deg, S);

  gate_kernel<<<1024, 256, 0, stream>>>(S, fused, out);
}